// Fuzzy_Attention_69200513073610
// MI455X (gfx1250) — compile-verified
//
#include <hip/hip_runtime.h>
#include <hip/hip_bf16.h>

// Problem constants (match reference: B=8, L=2048, D=512, R=512)
#define BB 8
#define LL 2048
#define DD 512
#define RR 512
#define MTOT (BB * LL)   // 16384 rows of q / z

typedef float v2f __attribute__((ext_vector_type(2)));
typedef float v8f __attribute__((ext_vector_type(8)));

// Native fp32 WMMA: D(16x16,f32) = A(16x4,f32) * B(4x16,f32) + C
__device__ __forceinline__ v8f wmma_f32(v2f a, v2f b, v8f c) {
  return __builtin_amdgcn_wmma_f32_16x16x4_f32(
      /*neg_a=*/false, a, /*neg_b=*/false, b,
      /*c_mod=*/(short)0, c, /*reuse_a=*/false, /*reuse_b=*/false);
}

__device__ __forceinline__ float waveRedSum(float v) {
#pragma unroll
  for (int m = 16; m >= 1; m >>= 1) v += __shfl_xor(v, m, 32);
  return v;
}
__device__ __forceinline__ float waveRedMax(float v) {
#pragma unroll
  for (int m = 16; m >= 1; m >>= 1) v = fmaxf(v, __shfl_xor(v, m, 32));
  return v;
}

// ---------------------------------------------------------------------------
// K2: per-rule precompute: iw = 1/w^2, cw = centers*iw, cc = sum(c^2*iw)
// One wave per rule row (R=512 rows). Block 256 -> 8 rows/block, 64 blocks.
// ---------------------------------------------------------------------------
__global__ __launch_bounds__(256) void prep_kernel(
    const float* __restrict__ centers, const float* __restrict__ widths,
    float* __restrict__ iw, float* __restrict__ cw, float* __restrict__ cc) {
  int row = (blockIdx.x * blockDim.x + threadIdx.x) >> 5;
  int lane = threadIdx.x & 31;
  if (row >= RR) return;
  const float* crow = centers + row * DD;
  const float* wrow = widths + row * DD;
  float acc = 0.0f;
#pragma unroll
  for (int t = 0; t < DD / 32; ++t) {
    int d = lane + 32 * t;
    float w = wrow[d];
    float inv = 1.0f / (w * w);
    float c = crow[d];
    iw[row * DD + d] = inv;
    cw[row * DD + d] = c * inv;
    acc += c * c * inv;
  }
  acc = waveRedSum(acc);
  if (lane == 0) cc[row] = acc;
}

// ---------------------------------------------------------------------------
// K1: q = query @ Wq^T + bq.  A:(M=16384,K=512) row-major, B:(N=512,K=512)
// row-major (so B[k][n] = Wq[n*K+k], contiguous per-lane v2f loads).
// Wave tile 16x64 (4 N-subtiles share the A fragment). Block = 8 waves
// stacked in M -> 128x64 per block. grid = (16384/128, 512/64) = (128, 8).
// ---------------------------------------------------------------------------
__global__ __launch_bounds__(256) void qproj_kernel(
    const float* __restrict__ A, const float* __restrict__ Bm,
    const float* __restrict__ bias, float* __restrict__ C) {
  const int K = DD, N = DD;
  int wave = threadIdx.x >> 5;
  int lane = threadIdx.x & 31;
  int hi = lane >> 4;       // 0: K=k0,k0+1   1: K=k0+2,k0+3
  int ln = lane & 15;
  int tile_m = blockIdx.x * 128 + wave * 16;
  int tile_n = blockIdx.y * 64;

  const float* arow = A + (tile_m + ln) * K + 2 * hi;
  const float* brow = Bm + (tile_n + ln) * K + 2 * hi;
  __builtin_prefetch(arow, 0, 1);  // global_prefetch_b8

  v8f c0 = {}, c1 = {}, c2 = {}, c3 = {};
  for (int k0 = 0; k0 < K; k0 += 4) {
    v2f a = *(const v2f*)(arow + k0);
    v2f b0 = *(const v2f*)(brow + 0 * 16 * K + k0);
    v2f b1 = *(const v2f*)(brow + 1 * 16 * K + k0);
    v2f b2 = *(const v2f*)(brow + 2 * 16 * K + k0);
    v2f b3 = *(const v2f*)(brow + 3 * 16 * K + k0);
    c0 = wmma_f32(a, b0, c0);
    c1 = wmma_f32(a, b1, c1);
    c2 = wmma_f32(a, b2, c2);
    c3 = wmma_f32(a, b3, c3);
  }
  float bq0 = bias[tile_n + 0 + ln], bq1 = bias[tile_n + 16 + ln];
  float bq2 = bias[tile_n + 32 + ln], bq3 = bias[tile_n + 48 + ln];
#pragma unroll
  for (int i = 0; i < 8; ++i) {
    int m = tile_m + i + 8 * hi;   // C layout: VGPR i -> M=i (lo) / i+8 (hi)
    float* crow = C + m * N + tile_n + ln;
    crow[0]  = c0[i] + bq0;
    crow[16] = c1[i] + bq1;
    crow[32] = c2[i] + bq2;
    crow[48] = c3[i] + bq3;
  }
}

// ---------------------------------------------------------------------------
// K3: z = -(0.5/D) * ( (q*q)@iw^T - 2*(q@cw^T) + cc ).
// One A-frag load of q, squared in registers -> two WMMA accumulators per
// N-subtile. Wave tile 16x32 (2 subtiles x {qq,qc}). Block 128x32.
// grid = (16384/128, 512/32) = (128, 16).
// ---------------------------------------------------------------------------
__global__ __launch_bounds__(256) void z_kernel(
    const float* __restrict__ q, const float* __restrict__ iw,
    const float* __restrict__ cw, const float* __restrict__ cc,
    float* __restrict__ z) {
  const int K = DD, N = RR;
  int wave = threadIdx.x >> 5;
  int lane = threadIdx.x & 31;
  int hi = lane >> 4;
  int ln = lane & 15;
  int tile_m = blockIdx.x * 128 + wave * 16;
  int tile_n = blockIdx.y * 32;

  const float* arow  = q  + (tile_m + ln) * K + 2 * hi;
  const float* iwrow = iw + (tile_n + ln) * K + 2 * hi;
  const float* cwrow = cw + (tile_n + ln) * K + 2 * hi;

  v8f qq0 = {}, qc0 = {}, qq1 = {}, qc1 = {};
  for (int k0 = 0; k0 < K; k0 += 4) {
    v2f a = *(const v2f*)(arow + k0);
    v2f asq = a * a;
    v2f bi0 = *(const v2f*)(iwrow + k0);
    v2f bi1 = *(const v2f*)(iwrow + 16 * K + k0);
    v2f bc0 = *(const v2f*)(cwrow + k0);
    v2f bc1 = *(const v2f*)(cwrow + 16 * K + k0);
    qq0 = wmma_f32(asq, bi0, qq0);
    qc0 = wmma_f32(a,   bc0, qc0);
    qq1 = wmma_f32(asq, bi1, qq1);
    qc1 = wmma_f32(a,   bc1, qc1);
  }
  const float scale = -0.5f / (float)DD;
  float cc0 = cc[tile_n + ln], cc1 = cc[tile_n + 16 + ln];
#pragma unroll
  for (int i = 0; i < 8; ++i) {
    int m = tile_m + i + 8 * hi;
    float* zrow = z + m * N + tile_n + ln;
    zrow[0]  = scale * (qq0[i] - 2.0f * qc0[i] + cc0);
    zrow[16] = scale * (qq1[i] - 2.0f * qc1[i] + cc1);
  }
}

// ---------------------------------------------------------------------------
// K4: row softmax over R=512, in place. One wave per row; 16 elems/lane.
// ---------------------------------------------------------------------------
__global__ __launch_bounds__(256) void softmax_kernel(float* __restrict__ z) {
  int row = (blockIdx.x * blockDim.x + threadIdx.x) >> 5;
  int lane = threadIdx.x & 31;
  if (row >= MTOT) return;
  float* p = z + (size_t)row * RR;
  float vals[RR / 32];
  float mx = -3.4e38f;
#pragma unroll
  for (int t = 0; t < RR / 32; ++t) {
    vals[t] = p[lane + 32 * t];
    mx = fmaxf(mx, vals[t]);
  }
  mx = waveRedMax(mx);
  float s = 0.0f;
#pragma unroll
  for (int t = 0; t < RR / 32; ++t) {
    vals[t] = __expf(vals[t] - mx);
    s += vals[t];
  }
  s = waveRedSum(s);
  float inv = 1.0f / s;
#pragma unroll
  for (int t = 0; t < RR / 32; ++t) p[lane + 32 * t] = vals[t] * inv;
}

// ---------------------------------------------------------------------------
// K5: conq[b] = Wc @ query[b] + bc  -> (R x D), K = L = 2048.
// A = Wc (R,L) contiguous-K; B[k][n] = query[b,k,n] (stride-D scalar loads,
// contiguous across lanes). Wave tile 16x64, block 128x64.
// grid = (512/128, 512/64, B) = (4, 8, 8). Bias is along M (rule index).
// ---------------------------------------------------------------------------
__global__ __launch_bounds__(256) void conq_kernel(
    const float* __restrict__ query, const float* __restrict__ Wc,
    const float* __restrict__ bc, float* __restrict__ conq) {
  const int K = LL, N = DD;
  int b = blockIdx.z;
  int wave = threadIdx.x >> 5;
  int lane = threadIdx.x & 31;
  int hi = lane >> 4;
  int ln = lane & 15;
  int tile_m = blockIdx.x * 128 + wave * 16;
  int tile_n = blockIdx.y * 64;

  const float* arow = Wc + (tile_m + ln) * K + 2 * hi;
  const float* Qb = query + b * LL * DD;

  v8f c0 = {}, c1 = {}, c2 = {}, c3 = {};
  for (int k0 = 0; k0 < K; k0 += 4) {
    int kk = k0 + 2 * hi;
    v2f a = *(const v2f*)(arow + k0);
    const float* bp = Qb + kk * DD + tile_n + ln;
    v2f b0, b1, b2, b3;
    b0.x = bp[0];       b0.y = bp[DD + 0];
    b1.x = bp[16];      b1.y = bp[DD + 16];
    b2.x = bp[32];      b2.y = bp[DD + 32];
    b3.x = bp[48];      b3.y = bp[DD + 48];
    c0 = wmma_f32(a, b0, c0);
    c1 = wmma_f32(a, b1, c1);
    c2 = wmma_f32(a, b2, c2);
    c3 = wmma_f32(a, b3, c3);
  }
  float* Cb = conq + b * RR * DD;
#pragma unroll
  for (int i = 0; i < 8; ++i) {
    int m = tile_m + i + 8 * hi;
    float bias = bc[m];                 // bias along rule (M) axis
    float* crow = Cb + m * N + tile_n + ln;
    crow[0]  = c0[i] + bias;
    crow[16] = c1[i] + bias;
    crow[32] = c2[i] + bias;
    crow[48] = c3[i] + bias;
  }
}

// ---------------------------------------------------------------------------
// K6: out[b] = Fss[b] @ conq[b]  -> (L x D), K = R = 512.
// A = Fss contiguous-K; B[k][n] = conq[b,k,n] stride-D.
// grid = (2048/128, 512/64, B) = (16, 8, 8).
// ---------------------------------------------------------------------------
__global__ __launch_bounds__(256) void out_kernel(
    const float* __restrict__ Fss, const float* __restrict__ conq,
    float* __restrict__ out) {
  const int K = RR, N = DD;
  int b = blockIdx.z;
  int wave = threadIdx.x >> 5;
  int lane = threadIdx.x & 31;
  int hi = lane >> 4;
  int ln = lane & 15;
  int tile_m = blockIdx.x * 128 + wave * 16;
  int tile_n = blockIdx.y * 64;

  const float* arow = Fss + b * LL * RR + (tile_m + ln) * K + 2 * hi;
  const float* Bb_ = conq + b * RR * DD;

  v8f c0 = {}, c1 = {}, c2 = {}, c3 = {};
  for (int k0 = 0; k0 < K; k0 += 4) {
    int kk = k0 + 2 * hi;
    v2f a = *(const v2f*)(arow + k0);
    const float* bp = Bb_ + kk * DD + tile_n + ln;
    v2f b0, b1, b2, b3;
    b0.x = bp[0];       b0.y = bp[DD + 0];
    b1.x = bp[16];      b1.y = bp[DD + 16];
    b2.x = bp[32];      b2.y = bp[DD + 32];
    b3.x = bp[48];      b3.y = bp[DD + 48];
    c0 = wmma_f32(a, b0, c0);
    c1 = wmma_f32(a, b1, c1);
    c2 = wmma_f32(a, b2, c2);
    c3 = wmma_f32(a, b3, c3);
  }
  float* Ob = out + b * LL * DD;
#pragma unroll
  for (int i = 0; i < 8; ++i) {
    int m = tile_m + i + 8 * hi;
    float* crow = Ob + m * N + tile_n + ln;
    crow[0]  = c0[i];
    crow[16] = c1[i];
    crow[32] = c2[i];
    crow[48] = c3[i];
  }
}

// ---------------------------------------------------------------------------
// Launch
// inputs (reference order): query, Wq, bq, Wc, bc, centers, widths
// ---------------------------------------------------------------------------
extern "C" void kernel_launch(void* const* d_in, const int* in_sizes, int n_in,
                              void* d_out, int out_size, void* d_ws, size_t ws_size,
                              hipStream_t stream) {
  const float* query   = (const float*)d_in[0];
  const float* Wq      = (const float*)d_in[1];
  const float* bq      = (const float*)d_in[2];
  const float* Wc      = (const float*)d_in[3];
  const float* bc      = (const float*)d_in[4];
  const float* centers = (const float*)d_in[5];
  const float* widths  = (const float*)d_in[6];
  float* out = (float*)d_out;

  // workspace layout (floats)
  float* q    = (float*)d_ws;          // MTOT*DD      =  8,388,608
  float* z    = q    + (size_t)MTOT * DD;   // MTOT*RR =  8,388,608
  float* iw   = z    + (size_t)MTOT * RR;   // RR*DD   =    262,144
  float* cw   = iw   + (size_t)RR * DD;     // RR*DD   =    262,144
  float* cc   = cw   + (size_t)RR * DD;     // RR      =        512
  float* conq = cc   + RR;                  // BB*RR*DD=  2,097,152
  // total ~74 MB

  // K2: rule precompute (independent of K1)
  prep_kernel<<<RR / 8, 256, 0, stream>>>(centers, widths, iw, cw, cc);

  // K1: q = query @ Wq^T + bq
  qproj_kernel<<<dim3(MTOT / 128, DD / 64), 256, 0, stream>>>(query, Wq, bq, q);

  // K3: fuzzy logits z
  z_kernel<<<dim3(MTOT / 128, RR / 32), 256, 0, stream>>>(q, iw, cw, cc, z);

  // K4: softmax along R (in place -> Fss)
  softmax_kernel<<<MTOT / 8, 256, 0, stream>>>(z);

  // K5: conq = Wc @ query[b] + bc  (independent of K1-K4, stream-ordered)
  conq_kernel<<<dim3(RR / 128, DD / 64, BB), 256, 0, stream>>>(query, Wc, bc, conq);

  // K6: out = Fss @ conq
  out_kernel<<<dim3(LL / 128, DD / 64, BB), 256, 0, stream>>>(z, conq, out);
}